// GATLayer_24558622998848
// MI455X (gfx1250) — compile-verified
//
#include <hip/hip_runtime.h>
#include <hip/hip_bf16.h>
#include <math.h>

#define IN_F   256
#define OUT_F  64
#define KCHUNK 64
#define CHUNK_FLOATS (KCHUNK * OUT_F)          // 4096 floats = 16 KB
#define CHUNK_BYTES  (CHUNK_FLOATS * 4)

typedef float v2f __attribute__((ext_vector_type(2)));
typedef float v8f __attribute__((ext_vector_type(8)));

// ---- monotone order-preserving float<->uint mapping (for atomic max) ----
__device__ __forceinline__ unsigned ford(float f) {
  unsigned u = __float_as_uint(f);
  return (u & 0x80000000u) ? ~u : (u | 0x80000000u);
}
__device__ __forceinline__ float funord(unsigned u) {
  return (u & 0x80000000u) ? __uint_as_float(u & 0x7FFFFFFFu) : __uint_as_float(~u);
}

// ---- kernel 0: init h, per-node running max, denom ----
__global__ void gat_init(float* __restrict__ h, unsigned* __restrict__ emax,
                         float* __restrict__ denom, int nh, int nn) {
  int i = blockIdx.x * blockDim.x + threadIdx.x;
  if (i < nh) h[i] = 0.0f;
  if (i < nn) { emax[i] = ford(-3.0e38f); denom[i] = 0.0f; }
}

// ---- kernel 1: z = X @ W  via V_WMMA_F32_16X16X4_F32 ----
// Block = 128 threads (4 waves). Block handles 16 rows of X; wave w handles
// output cols [16w, 16w+16). W staged in double-buffered LDS chunks via
// async global->LDS DMA (GLOBAL_LOAD_ASYNC_TO_LDS_B128, ASYNCcnt-tracked),
// so the next chunk's copy overlaps this chunk's WMMA loop.
__global__ __launch_bounds__(128) void gat_gemm_wmma(
    const float* __restrict__ A,   // [M, 256]
    const float* __restrict__ W,   // [256, 64]
    float* __restrict__ Z,         // [M, 64]
    int M) {
  __shared__ float Wlds[2][CHUNK_FLOATS];   // 32 KB, double buffered (LDS offset 0)

  // Escape hatch: the async-to-LDS copies below address Wlds only via raw
  // integer LDS offsets inside inline asm. Capture the flat pointer in an
  // opaque asm so alias analysis must treat Wlds as written by the asm
  // memory clobbers (otherwise the B-fragment loads fold to undef).
  {
    float* wesc = &Wlds[0][0];
    asm volatile("" : "+v"(wesc) :: "memory");
  }

  const int tid   = threadIdx.x;
  const int lane  = tid & 31;
  const int wave  = tid >> 5;              // 0..3 -> N tile
  const int hi    = lane >> 4;             // lane group 0 / 1
  const int l16   = lane & 15;
  const int mbase = blockIdx.x * 16;
  const int ncol  = wave * 16;
  const int bcol  = ncol + l16;

  // clamp row so all lanes stay active (EXEC must be all-1s for WMMA)
  int row = mbase + l16;
  if (row >= M) row = M - 1;
  const float* arow = A + (size_t)row * IN_F;

  // --- async-stage chunk 0 into buffer 0 ---
  {
    const float4* srcg = (const float4*)W;          // chunk 0 = W[0:64, :]
    #pragma unroll
    for (int i = 0; i < 8; ++i) {
      int idx = tid + 128 * i;                       // float4 index in chunk
      unsigned long long ga = (unsigned long long)(srcg + idx);
      unsigned ldsb = (unsigned)(idx * 16);          // byte offset in buffer 0
      asm volatile("global_load_async_to_lds_b128 %0, %1, off"
                   :: "v"(ldsb), "v"(ga) : "memory");
    }
  }
  asm volatile("s_wait_asynccnt 0x0" ::: "memory");
  __syncthreads();

  v8f c = {};  // accumulator

  const int NCHUNK = IN_F / KCHUNK;  // 4
  for (int ch = 0; ch < NCHUNK; ++ch) {
    const int   k0  = ch * KCHUNK;
    const int   buf = ch & 1;
    const float* Wb = Wlds[buf];

    // --- kick off async copy of next chunk into the other buffer ---
    if (ch + 1 < NCHUNK) {
      const float4* srcg = (const float4*)(W + (size_t)(k0 + KCHUNK) * OUT_F);
      unsigned base = (unsigned)(((ch + 1) & 1) * CHUNK_BYTES);
      #pragma unroll
      for (int i = 0; i < 8; ++i) {
        int idx = tid + 128 * i;
        unsigned long long ga = (unsigned long long)(srcg + idx);
        unsigned ldsb = base + (unsigned)(idx * 16);
        asm volatile("global_load_async_to_lds_b128 %0, %1, off"
                     :: "v"(ldsb), "v"(ga) : "memory");
      }
    }

    // --- software-pipelined WMMA loop over this chunk ---
    // A frag (16x4 f32): lanes 0-15 K={0,1}, lanes 16-31 K={2,3}
    // B frag (4x16 f32): VGPR j: lanes 0-15 -> K=j, lanes 16-31 -> K=2+j
    v2f a_cur = *(const v2f*)(arow + k0 + 2 * hi);
    v2f b_cur;
    b_cur.x = Wb[(2 * hi + 0) * OUT_F + bcol];
    b_cur.y = Wb[(2 * hi + 1) * OUT_F + bcol];

    #pragma unroll
    for (int kk = 0; kk < KCHUNK; kk += 4) {
      v2f a_nxt = a_cur, b_nxt = b_cur;
      if (kk + 4 < KCHUNK) {               // compile-time under full unroll
        a_nxt = *(const v2f*)(arow + k0 + kk + 4 + 2 * hi);
        b_nxt.x = Wb[(kk + 4 + 2 * hi) * OUT_F + bcol];
        b_nxt.y = Wb[(kk + 5 + 2 * hi) * OUT_F + bcol];
      }
      c = __builtin_amdgcn_wmma_f32_16x16x4_f32(
              /*neg_a=*/false, a_cur, /*neg_b=*/false, b_cur,
              /*c_mod=*/(short)0, c, /*reuse_a=*/false, /*reuse_b=*/false);
      a_cur = a_nxt;
      b_cur = b_nxt;
    }

    // next chunk's async copy must land; barrier also protects buffer reuse
    asm volatile("s_wait_asynccnt 0x0" ::: "memory");
    __syncthreads();
  }

  // D layout: VGPR r, lanes 0-15 -> (M=r, N=l16); lanes 16-31 -> (M=8+r, N=l16)
  const int outrow = mbase + 8 * hi;
  const int outcol = ncol + l16;
  #pragma unroll
  for (int r = 0; r < 8; ++r) {
    int rr = outrow + r;
    if (rr < M) Z[(size_t)rr * OUT_F + outcol] = c[r];
  }
}

// ---- kernel 2: per-node scores s_src = z.Wa[:64], s_dst = z.Wa[64:] ----
__global__ __launch_bounds__(256) void gat_scores(
    const float* __restrict__ Z, const float* __restrict__ Wa,
    float* __restrict__ s_src, float* __restrict__ s_dst, int N) {
  int warp = (blockIdx.x * blockDim.x + threadIdx.x) >> 5;
  int lane = threadIdx.x & 31;
  if (warp >= N) return;
  float z0 = Z[(size_t)warp * OUT_F + lane];
  float z1 = Z[(size_t)warp * OUT_F + 32 + lane];
  float ps = z0 * Wa[lane]      + z1 * Wa[32 + lane];
  float pd = z0 * Wa[64 + lane] + z1 * Wa[96 + lane];
  #pragma unroll
  for (int off = 16; off > 0; off >>= 1) {
    ps += __shfl_xor(ps, off, 32);
    pd += __shfl_xor(pd, off, 32);
  }
  if (lane == 0) { s_src[warp] = ps; s_dst[warp] = pd; }
}

// ---- kernel 3: e = leaky_relu(s_src[src] + s_dst[dst]); segment max ----
__global__ void gat_edge_logits(const int* __restrict__ src, const int* __restrict__ dst,
                                const float* __restrict__ s_src, const float* __restrict__ s_dst,
                                float* __restrict__ ebuf, unsigned* __restrict__ emax, int E) {
  int i = blockIdx.x * blockDim.x + threadIdx.x;
  if (i >= E) return;
  float e = s_src[src[i]] + s_dst[dst[i]];
  e = (e > 0.0f) ? e : 0.2f * e;
  ebuf[i] = e;
  atomicMax(&emax[dst[i]], ford(e));
}

// ---- kernel 4: ex = exp(e - max[dst]); segment sum ----
__global__ void gat_edge_exp(const int* __restrict__ dst, float* __restrict__ ebuf,
                             const unsigned* __restrict__ emax, float* __restrict__ denom, int E) {
  int i = blockIdx.x * blockDim.x + threadIdx.x;
  if (i >= E) return;
  int d = dst[i];
  float ex = __expf(ebuf[i] - funord(emax[d]));
  ebuf[i] = ex;
  atomicAdd(&denom[d], ex);
}

// ---- kernel 5: h[dst] += (ex/denom[dst]) * z[src], wave per edge ----
__global__ __launch_bounds__(256) void gat_scatter(
    const int* __restrict__ src, const int* __restrict__ dst,
    const float* __restrict__ ebuf, const float* __restrict__ denom,
    const float* __restrict__ Z, float* __restrict__ H, int E) {
  int warp = (blockIdx.x * blockDim.x + threadIdx.x) >> 5;
  int lane = threadIdx.x & 31;
  if (warp >= E) return;
  int s = src[warp], d = dst[warp];
  float alpha = ebuf[warp] / denom[d];
  float v0 = alpha * Z[(size_t)s * OUT_F + lane];
  float v1 = alpha * Z[(size_t)s * OUT_F + 32 + lane];
  atomicAdd(&H[(size_t)d * OUT_F + lane], v0);
  atomicAdd(&H[(size_t)d * OUT_F + 32 + lane], v1);
}

extern "C" void kernel_launch(void* const* d_in, const int* in_sizes, int n_in,
                              void* d_out, int out_size, void* d_ws, size_t ws_size,
                              hipStream_t stream) {
  const float* features = (const float*)d_in[0];
  const int*   edge_src = (const int*)d_in[1];
  const int*   edge_dst = (const int*)d_in[2];
  const float* W_fc     = (const float*)d_in[3];
  const float* W_attn   = (const float*)d_in[4];
  float* H = (float*)d_out;

  const int N = in_sizes[0] / IN_F;  // 50000
  const int E = in_sizes[1];         // 800000

  // workspace layout (floats)
  float*    Z     = (float*)d_ws;                 // N*64
  float*    s_src = Z + (size_t)N * OUT_F;        // N
  float*    s_dst = s_src + N;                    // N
  float*    ebuf  = s_dst + N;                    // E
  unsigned* emax  = (unsigned*)(ebuf + E);        // N
  float*    denom = (float*)(emax + N);           // N

  const int nh = N * OUT_F;
  gat_init<<<(nh + 255) / 256, 256, 0, stream>>>(H, emax, denom, nh, N);

  gat_gemm_wmma<<<(N + 15) / 16, 128, 0, stream>>>(features, W_fc, Z, N);

  gat_scores<<<(int)(((size_t)N * 32 + 255) / 256), 256, 0, stream>>>(Z, W_attn, s_src, s_dst, N);

  gat_edge_logits<<<(E + 255) / 256, 256, 0, stream>>>(edge_src, edge_dst, s_src, s_dst,
                                                       ebuf, emax, E);

  gat_edge_exp<<<(E + 255) / 256, 256, 0, stream>>>(edge_dst, ebuf, emax, denom, E);

  gat_scatter<<<(int)(((size_t)E * 32 + 255) / 256), 256, 0, stream>>>(edge_src, edge_dst,
                                                                       ebuf, denom, Z, H, E);
}